// DifferentSoftQNetwork_87737591923434
// MI455X (gfx1250) — compile-verified
//
#include <hip/hip_runtime.h>

typedef __attribute__((ext_vector_type(2))) float v2f;
typedef __attribute__((ext_vector_type(8))) float v8f;

#define NOPT   8
#define BATCH  2048
#define DIN    160     // 128 state + 32 action
#define SH     512     // STATE_HIDDEN
#define HID    64      // HIDDEN
#define XS_STRIDE 162  // padded LDS strides (even -> b64-alignable, conflict-avoiding)
#define H1_STRIDE 514
#define H2_STRIDE 66

// ---------------- kernel A: zero option counters ----------------
__global__ void qnet_zero_counts(int* counts) {
    if (threadIdx.x < NOPT) counts[threadIdx.x] = 0;
}

// ---------------- kernel B: bucket samples by option ----------------
__global__ void qnet_bucket(const int* __restrict__ option,
                            int* __restrict__ counts,
                            int* __restrict__ idxlist) {
    int b = blockIdx.x * blockDim.x + threadIdx.x;
    if (b < BATCH) {
        int o = option[b];
        int pos = atomicAdd(&counts[o], 1);
        idxlist[o * BATCH + pos] = b;
    }
}

// ---------------- kernel C: grouped GEMM MLP via fp32 WMMA ----------------
__global__ __launch_bounds__(256)
void qnet_mlp(const float* __restrict__ state,
              const float* __restrict__ action,
              const int*   __restrict__ counts,
              const int*   __restrict__ idxlist,
              const float* __restrict__ W1,   // [8][160][512]
              const float* __restrict__ W2,   // [8][512][64]
              const float* __restrict__ W3,   // [8][64]
              float*       __restrict__ out)  // [2048]
{
    const int o = blockIdx.x;          // option 0..7
    const int t = blockIdx.y;          // tile 0..127 within this option's bucket
    const int cnt = counts[o];
    if (t * 16 >= cnt) return;
    const int valid = (cnt - t * 16 < 16) ? (cnt - t * 16) : 16;

    __shared__ float Xs [16 * XS_STRIDE];   // gathered inputs, 16 x 160
    __shared__ float H1s[16 * H1_STRIDE];   // layer-1 activations, 16 x 512
    __shared__ float H2s[16 * H2_STRIDE];   // layer-2 activations, 16 x 64
    __shared__ float w3s[HID];
    __shared__ int   rows[16];

    const int tid = threadIdx.x;

    if (tid < 16)
        rows[tid] = (tid < valid) ? idxlist[o * BATCH + t * 16 + tid] : -1;
    if (tid < HID)
        w3s[tid] = W3[o * HID + tid];
    __syncthreads();

    // Gather X = [state || action]; zero-fill rows past `valid` (zeros are
    // relu-safe and produce out=0 which is never stored).
    for (int i = tid; i < 16 * DIN; i += 256) {
        int r = i / DIN, c = i % DIN;
        int b = rows[r];
        float v = 0.f;
        if (b >= 0)
            v = (c < 128) ? state[(size_t)b * 128 + c]
                          : action[(size_t)b * 32 + (c - 128)];
        Xs[r * XS_STRIDE + c] = v;
    }
    __syncthreads();

    const int wave  = tid >> 5;
    const int lane  = tid & 31;
    const int mrow  = lane & 15;          // A-frag row / B-frag column / C column
    const int khalf = (lane >> 4) << 1;   // 0 for lanes 0-15, 2 for lanes 16-31
    const int crow  = (lane >> 4) << 3;   // C/D: M base = 0 or 8

    // ---------- Layer 1: (16x160) x (160x512), 8 waves x 4 N-tiles ----------
    const float* W1o = W1 + (size_t)o * DIN * SH;
    for (int j = 0; j < 4; ++j) {
        const int n = (wave * 4 + j) * 16;
        v8f acc = {};
        for (int k = 0; k < DIN; k += 4) {
            v2f a, bf;
            const float* ap = &Xs[mrow * XS_STRIDE + k + khalf];
            a.x = ap[0];
            a.y = ap[1];
            const float* bp = W1o + (size_t)(k + khalf) * SH + n + mrow;
            bf.x = bp[0];
            bf.y = bp[SH];
            acc = __builtin_amdgcn_wmma_f32_16x16x4_f32(
                      false, a, false, bf, (short)0, acc, false, false);
        }
        #pragma unroll
        for (int i = 0; i < 8; ++i) {
            float v = acc[i];
            H1s[(crow + i) * H1_STRIDE + n + mrow] = (v > 0.f) ? v : 0.f;
        }
    }
    __syncthreads();

    // ---------- Layer 2: (16x512) x (512x64), waves 0..3, 1 N-tile each ----------
    const float* W2o = W2 + (size_t)o * SH * HID;
    if (wave < 4) {                       // wave-uniform branch: EXEC all-ones inside
        const int n = wave * 16;
        v8f acc = {};
        for (int k = 0; k < SH; k += 4) {
            v2f a, bf;
            const float* ap = &H1s[mrow * H1_STRIDE + k + khalf];
            a.x = ap[0];
            a.y = ap[1];
            const float* bp = W2o + (size_t)(k + khalf) * HID + n + mrow;
            bf.x = bp[0];
            bf.y = bp[HID];
            acc = __builtin_amdgcn_wmma_f32_16x16x4_f32(
                      false, a, false, bf, (short)0, acc, false, false);
        }
        #pragma unroll
        for (int i = 0; i < 8; ++i) {
            float v = acc[i];
            H2s[(crow + i) * H2_STRIDE + n + mrow] = (v > 0.f) ? v : 0.f;
        }
    }
    __syncthreads();

    // ---------- Layer 3: (16x64) x (64x1), lanes 0..15 of wave 0 ----------
    if (wave == 0 && lane < valid) {
        float acc = 0.f;
        #pragma unroll
        for (int k = 0; k < HID; ++k)
            acc += H2s[lane * H2_STRIDE + k] * w3s[k];
        out[rows[lane]] = acc;
    }
}

extern "C" void kernel_launch(void* const* d_in, const int* in_sizes, int n_in,
                              void* d_out, int out_size, void* d_ws, size_t ws_size,
                              hipStream_t stream) {
    const float* state  = (const float*)d_in[0];
    const int*   option = (const int*)  d_in[1];
    const float* action = (const float*)d_in[2];
    const float* W1     = (const float*)d_in[3];
    const float* W2     = (const float*)d_in[4];
    const float* W3     = (const float*)d_in[5];
    float*       out    = (float*)d_out;

    int* counts  = (int*)d_ws;         // 8 ints
    int* idxlist = counts + NOPT;      // 8 * 2048 ints

    qnet_zero_counts<<<1, 32, 0, stream>>>(counts);
    qnet_bucket<<<BATCH / 256, 256, 0, stream>>>(option, counts, idxlist);

    dim3 grid(NOPT, BATCH / 16);       // worst case: one option owns all samples
    qnet_mlp<<<grid, 256, 0, stream>>>(state, action, counts, idxlist,
                                       W1, W2, W3, out);
}